// GATConv_48395691491891
// MI455X (gfx1250) — compile-verified
//
#include <hip/hip_runtime.h>
#include <hip/hip_bf16.h>
#include <math.h>

#define IN_DIM   128
#define HEADS    4
#define OUT_DIM  32
#define HF       128   // HEADS*OUT_DIM
#define NEG_SLOPE 0.2f

typedef __attribute__((ext_vector_type(2))) float v2f;
typedef __attribute__((ext_vector_type(8))) float v8f;

// ---- monotone uint mapping for float atomic-max ------------------------------
__device__ __forceinline__ unsigned int f2ord(float f) {
    unsigned int b = __float_as_uint(f);
    return (b & 0x80000000u) ? ~b : (b | 0x80000000u);
}
__device__ __forceinline__ float ord2f(unsigned int u) {
    return (u & 0x80000000u) ? __uint_as_float(u ^ 0x80000000u) : __uint_as_float(~u);
}
#define ORD_NEG_INF 0x007FFFFFu   // f2ord(-inf)

__device__ __forceinline__ float leaky(float v) {
    return v > 0.0f ? v : NEG_SLOPE * v;
}

// ---- init: out = bias broadcast ---------------------------------------------
__global__ void gat_init_out(float* __restrict__ out, const float* __restrict__ bias, int total) {
    int i = blockIdx.x * blockDim.x + threadIdx.x;
    if (i < total) out[i] = bias[i & (HF - 1)];
}

// ---- init: segmax = -inf (ordered), denom = 0 -------------------------------
__global__ void gat_init_seg(unsigned int* __restrict__ segmax_u, float* __restrict__ denom, int total) {
    int i = blockIdx.x * blockDim.x + threadIdx.x;
    if (i < total) { segmax_u[i] = ORD_NEG_INF; denom[i] = 0.0f; }
}

// ---- Wh = x @ W via V_WMMA_F32_16X16X4_F32 ----------------------------------
// Block = 256 threads = 8 waves. blockIdx.x selects a 16-row strip of x;
// wave w computes output columns [16w, 16w+16). K=128 -> 32 wmma steps of K=4.
__global__ void gat_gemm_wmma(const float* __restrict__ x, const float* __restrict__ W,
                              float* __restrict__ Wh, int N) {
    const int wave = threadIdx.x >> 5;          // 0..7 -> column tile
    const int lane = threadIdx.x & 31;
    const int l15  = lane & 15;
    const int half = lane >> 4;                 // 0: K+{0,1}; 1: K+{2,3}
    const int row0 = blockIdx.x * 16;
    const int col0 = wave * 16;

    // A operand row for this lane (clamp for a possible ragged tail strip; EXEC stays full)
    int arow = row0 + l15;
    arow = arow < N ? arow : N - 1;
    const float* __restrict__ xrow = x + (size_t)arow * IN_DIM;

    v8f c = {};
#pragma unroll
    for (int k0 = 0; k0 < IN_DIM; k0 += 4) {
        const int ka = k0 + half * 2;
        v2f a;                                   // A 16x4: V0=K ka, V1=K ka+1 (contiguous b64)
        a.x = xrow[ka];
        a.y = xrow[ka + 1];
        v2f b;                                   // B 4x16: rows striped across lanes
        b.x = W[(size_t)ka * HF + col0 + l15];
        b.y = W[(size_t)(ka + 1) * HF + col0 + l15];
        c = __builtin_amdgcn_wmma_f32_16x16x4_f32(false, a, false, b, (short)0, c, false, false);
    }

    // C/D 16x16 f32: VGPR v -> M = v + 8*half, N = l15
#pragma unroll
    for (int v = 0; v < 8; ++v) {
        const int m = v + half * 8;
        if (row0 + m < N) Wh[(size_t)(row0 + m) * HF + col0 + l15] = c[v];
    }
}

// ---- per-node attention halves: a_src/a_dst [N,H] ---------------------------
__global__ void gat_scores(const float* __restrict__ Wh, const float* __restrict__ att,
                           float* __restrict__ a_src, float* __restrict__ a_dst, int N) {
    int idx = blockIdx.x * blockDim.x + threadIdx.x;
    if (idx >= N * HEADS) return;
    const int n = idx >> 2, h = idx & 3;
    const float* __restrict__ w  = Wh  + (size_t)n * HF + h * OUT_DIM;
    const float* __restrict__ as = att + h * (2 * OUT_DIM);
    float ss = 0.0f, sd = 0.0f;
#pragma unroll
    for (int f = 0; f < OUT_DIM; ++f) {
        const float xv = w[f];
        ss += xv * as[f];
        sd += xv * as[OUT_DIM + f];
    }
    a_src[idx] = ss;
    a_dst[idx] = sd;
}

// ---- pass 1: segment max over dst (atomic max on ordered uint) --------------
__global__ void gat_edge_max(const long long* __restrict__ ei,
                             const float* __restrict__ a_src, const float* __restrict__ a_dst,
                             unsigned int* __restrict__ segmax_u, int E) {
    int idx = blockIdx.x * blockDim.x + threadIdx.x;
    if (idx >= E * HEADS) return;
    const int e = idx >> 2, h = idx & 3;
    const long long s = ei[e], d = ei[E + e];
    const float v = leaky(a_src[s * HEADS + h] + a_dst[d * HEADS + h]);
    atomicMax(&segmax_u[d * HEADS + h], f2ord(v));
}

// ---- pass 2: denom = segment sum of exp(logit - segmax) ---------------------
__global__ void gat_edge_denom(const long long* __restrict__ ei,
                               const float* __restrict__ a_src, const float* __restrict__ a_dst,
                               const unsigned int* __restrict__ segmax_u,
                               float* __restrict__ denom, int E) {
    int idx = blockIdx.x * blockDim.x + threadIdx.x;
    if (idx >= E * HEADS) return;
    const int e = idx >> 2, h = idx & 3;
    const long long s = ei[e], d = ei[E + e];
    const float v = leaky(a_src[s * HEADS + h] + a_dst[d * HEADS + h]);
    const float m = ord2f(segmax_u[d * HEADS + h]);
    atomicAdd(&denom[d * HEADS + h], __expf(v - m));
}

// ---- pass 3: out[dst] += alpha * Wh[src] (32 slots/edge, 4 feats/slot) ------
__global__ void gat_edge_scatter(const long long* __restrict__ ei,
                                 const float* __restrict__ a_src, const float* __restrict__ a_dst,
                                 const unsigned int* __restrict__ segmax_u,
                                 const float* __restrict__ denom,
                                 const float* __restrict__ Wh,
                                 float* __restrict__ out, int E) {
    const long long idx = (long long)blockIdx.x * blockDim.x + threadIdx.x;
    if (idx >= (long long)E * 32) return;
    const int e    = (int)(idx >> 5);
    const int slot = (int)(idx & 31);         // features [4*slot, 4*slot+4)
    const int h    = slot >> 3;               // head = feature/32
    const long long s = ei[e], d = ei[E + e];

    const float* __restrict__ wsrc = Wh + (size_t)s * HF + slot * 4;
    __builtin_prefetch(wsrc, 0, 0);           // global_prefetch_b8

    float v = leaky(a_src[s * HEADS + h] + a_dst[d * HEADS + h]);
    const float m  = ord2f(segmax_u[d * HEADS + h]);
    float dn = denom[d * HEADS + h];
    dn = dn > 1e-9f ? dn : 1e-9f;
    const float alpha = __expf(v - m) / dn;

    const float4 wv = *(const float4*)wsrc;   // global_load_b128
    float* __restrict__ op = out + (size_t)d * HF + slot * 4;
    atomicAdd(op + 0, alpha * wv.x);
    atomicAdd(op + 1, alpha * wv.y);
    atomicAdd(op + 2, alpha * wv.z);
    atomicAdd(op + 3, alpha * wv.w);
}

extern "C" void kernel_launch(void* const* d_in, const int* in_sizes, int n_in,
                              void* d_out, int out_size, void* d_ws, size_t ws_size,
                              hipStream_t stream) {
    const float*     x    = (const float*)d_in[0];
    const long long* ei   = (const long long*)d_in[1];
    const float*     W    = (const float*)d_in[2];
    const float*     att  = (const float*)d_in[3];
    const float*     bias = (const float*)d_in[4];
    float*           out  = (float*)d_out;

    const int N = in_sizes[0] / IN_DIM;
    const int E = in_sizes[1] / 2;

    // workspace layout (floats): Wh | a_src | a_dst | segmax_u | denom
    float*        Wh       = (float*)d_ws;
    float*        a_src    = Wh + (size_t)N * HF;
    float*        a_dst    = a_src + (size_t)N * HEADS;
    unsigned int* segmax_u = (unsigned int*)(a_dst + (size_t)N * HEADS);
    float*        denom    = (float*)(segmax_u + (size_t)N * HEADS);

    const int T = 256;
    gat_init_out<<<(N * HF + T - 1) / T, T, 0, stream>>>(out, bias, N * HF);
    gat_init_seg<<<(N * HEADS + T - 1) / T, T, 0, stream>>>(segmax_u, denom, N * HEADS);
    gat_gemm_wmma<<<(N + 15) / 16, T, 0, stream>>>(x, W, Wh, N);
    gat_scores<<<(N * HEADS + T - 1) / T, T, 0, stream>>>(Wh, att, a_src, a_dst, N);
    gat_edge_max<<<(E * HEADS + T - 1) / T, T, 0, stream>>>(ei, a_src, a_dst, segmax_u, E);
    gat_edge_denom<<<(E * HEADS + T - 1) / T, T, 0, stream>>>(ei, a_src, a_dst, segmax_u, denom, E);
    const long long scatterThreads = (long long)E * 32;
    gat_edge_scatter<<<(int)((scatterThreads + T - 1) / T), T, 0, stream>>>(
        ei, a_src, a_dst, segmax_u, denom, Wh, out, E);
}